// MultiComponentEntropy_16569983828644
// MI455X (gfx1250) — compile-verified
//
#include <hip/hip_runtime.h>
#include <hip/hip_bf16.h>

typedef float v2f __attribute__((ext_vector_type(2)));
typedef float v8f __attribute__((ext_vector_type(8)));

#define B_DIM 32
#define L_DIM 4096
#define C_DIM 512
#define SF 16
#define TILE_ELEMS (B_DIM * SF * SF)   /* 8192 */
#define IN_DIM 8271
#define K1PAD 8272                     /* IN_DIM padded to multiple of 4 */
#define H1_DIM 256
#define H2_DIM 128
#define H3_DIM 64
#define NUM_MC 10
#define TWO_PI_E 17.079468445347134f

// ---------------- hash-based Gaussian noise (Box-Muller) ----------------
__device__ __forceinline__ unsigned hash_u32(unsigned x) {
    x ^= x >> 16; x *= 0x7feb352dU;
    x ^= x >> 15; x *= 0x846ca68bU;
    x ^= x >> 16;
    return x;
}
__device__ __forceinline__ float hnorm(unsigned s) {
    unsigned a = hash_u32(s ^ 0x9E3779B9u);
    unsigned b = hash_u32(a + 0x7F4A7C15u + s);
    float u1 = (float)((a >> 8) + 1u) * (1.0f / 16777216.0f);  // (0,1]
    float u2 = (float)(b >> 8) * (1.0f / 16777216.0f);
    return sqrtf(-2.0f * logf(u1)) * cosf(6.28318530718f * u2);
}

// ---------------- ordered-bits mapping for exact lower median ----------------
__device__ __forceinline__ unsigned f2ord(float f) {
    unsigned u = __float_as_uint(f);
    return (u & 0x80000000u) ? ~u : (u | 0x80000000u);
}
__device__ __forceinline__ float ord2f(unsigned u) {
    unsigned v = (u & 0x80000000u) ? (u & 0x7FFFFFFFu) : ~u;
    return __uint_as_float(v);
}

// generic->LDS byte offset (flat LDS aperture address keeps the LDS offset in
// the low 32 bits; hardware adds the wave's LDS_BASE)
__device__ __forceinline__ unsigned lds_off(const volatile void* p) {
    return (unsigned)(unsigned long long)p;
}

// ---------------- block reductions (256 threads) ----------------
__device__ float block_sum(float v, float* sred) {
    int tid = threadIdx.x;
    __syncthreads();
    sred[tid] = v;
    __syncthreads();
    for (int s = 128; s > 0; s >>= 1) {
        if (tid < s) sred[tid] += sred[tid + s];
        __syncthreads();
    }
    return sred[0];
}
__device__ float block_min(float v, float* sred) {
    int tid = threadIdx.x;
    __syncthreads();
    sred[tid] = v;
    __syncthreads();
    for (int s = 128; s > 0; s >>= 1) {
        if (tid < s) sred[tid] = fminf(sred[tid], sred[tid + s]);
        __syncthreads();
    }
    return sred[0];
}
__device__ float block_max(float v, float* sred) {
    int tid = threadIdx.x;
    __syncthreads();
    sred[tid] = v;
    __syncthreads();
    for (int s = 128; s > 0; s >>= 1) {
        if (tid < s) sred[tid] = fmaxf(sred[tid], sred[tid + s]);
        __syncthreads();
    }
    return sred[0];
}

// ---------------- per-node feature + entropy-component kernel ----------------
__global__ void __launch_bounds__(256)
features_kernel(const float* __restrict__ z,
                const float* __restrict__ eh, int Neh,
                const int* __restrict__ t_idx,
                const int* __restrict__ f_idx,
                const int* __restrict__ kptr,
                float* __restrict__ feat,
                float* __restrict__ comps) {
    __shared__ float s_z[TILE_ELEMS];
    __shared__ float sred[256];
    __shared__ float twc[16], tws[16];

    const int n = blockIdx.x;
    const int tid = threadIdx.x;
    const int t = t_idx[n];
    const int f = f_idx[n];

    if (tid < 16) {
        float ang = -6.28318530718f * (float)tid / 16.0f;
        twc[tid] = cosf(ang);
        tws[tid] = sinf(ang);
    }

    // ---- async-copy the local (B, SF, SF) tile straight into LDS ----
    // CDNA5 GLOBAL_LOAD_ASYNC_TO_LDS_B32: each lane copies one dword from global memory to its LDS offset;
    // tracked by ASYNCcnt (peaks at 32 outstanding per wave, limit 63).
    for (int idx = tid; idx < TILE_ELEMS; idx += 256) {
        int b = idx >> 8;
        int r = (idx >> 4) & 15;
        int c = idx & 15;
        const float* gp =
            &z[((size_t)b * L_DIM + (size_t)t * SF + r) * C_DIM + (size_t)f * SF + c];
        unsigned lo = lds_off(&s_z[idx]);
        asm volatile("global_load_async_to_lds_b32 %0, %1, off"
                     :: "v"(lo), "v"(gp) : "memory");
    }
    asm volatile("s_wait_asynccnt 0x0" ::: "memory");
    __syncthreads();

    // local accumulations over this thread's elements (+ feature-row store)
    float lsum = 0.f, lsq = 0.f, labs = 0.f, lpos = 0.f, lneg = 0.f;
    float lmin = 3.4e38f, lmax = -3.4e38f;
    float l_score = 0.f, l_d12 = 0.f, l_d23 = 0.f;
    for (int idx = tid; idx < TILE_ELEMS; idx += 256) {
        float v = s_z[idx];
        feat[(size_t)n * K1PAD + idx] = v;
        lsum += v; lsq += v * v; labs += fabsf(v);
        lpos += (v > 0.f) ? 1.f : 0.f;
        lneg += (v < 0.f) ? 1.f : 0.f;
        lmin = fminf(lmin, v); lmax = fmaxf(lmax, v);

        unsigned base = ((unsigned)n * 0x01000193u) ^ (unsigned)idx;
        // score entropy: unbiased variance of 10 MC noise draws (z cancels)
        float s = 0.f, s2 = 0.f;
        for (int m = 0; m < NUM_MC; ++m) {
            float nv = 0.1f * hnorm(base * 16u + (unsigned)m * 0x1003u + 0xA5u);
            s += nv; s2 += nv * nv;
        }
        float mu = s * (1.f / NUM_MC);
        float var = (s2 - (float)NUM_MC * mu * mu) * (1.f / (NUM_MC - 1));
        l_score += 0.5f * logf(TWO_PI_E * (var + 1e-8f));
        // solver KL: only noise differences survive (z cancels in mu1-mu2)
        float e1 = 0.10f * hnorm(base ^ 0x11111111u);
        float e2 = 0.05f * hnorm(base ^ 0x22222222u);
        float e3 = 0.02f * hnorm(base ^ 0x33333333u);
        l_d12 += (e1 - e2) * (e1 - e2);
        l_d23 += (e2 - e3) * (e2 - e3);
    }

    float vsum = block_sum(lsum, sred);
    float vsq  = block_sum(lsq, sred);
    float sabs = block_sum(labs, sred);
    float npos = block_sum(lpos, sred);
    float nneg = block_sum(lneg, sred);
    float vmin = block_min(lmin, sred);
    float vmax = block_max(lmax, sred);
    float score_sum = block_sum(l_score, sred);
    float d12 = block_sum(l_d12, sred);
    float d23 = block_sum(l_d23, sred);

    // spectral entropy: 16-point DFT along local time axis per (b, j) column
    float l_spec = 0.f;
    for (int c = tid; c < B_DIM * SF; c += 256) {
        int b = c >> 4, j = c & 15;
        float ps[16];
        float tot = 0.f;
        for (int w = 0; w < 16; ++w) {
            float re = 0.f, im = 0.f;
            for (int i = 0; i < 16; ++i) {
                float x = s_z[b * 256 + i * 16 + j];
                int p = (w * i) & 15;
                re += x * twc[p];
                im += x * tws[p];
            }
            ps[w] = re * re + im * im;
            tot += ps[w];
        }
        float inv = 1.f / (tot + 1e-8f);
        for (int w = 0; w < 16; ++w) {
            float p = ps[w] * inv;
            l_spec -= p * logf(p + 1e-8f);
        }
    }
    float h_spec = block_sum(l_spec, sred);

    // temporal entropy (same for all nodes; cheap, recompute per block)
    float lt = 0.f, lt2 = 0.f;
    int M3 = 3 * Neh;
    for (int idx = tid; idx < M3; idx += 256) {
        float d = eh[idx + Neh] - eh[idx];
        lt += d; lt2 += d * d;
    }
    float ts = block_sum(lt, sred);
    float ts2 = block_sum(lt2, sred);
    float tmean = ts / (float)M3;
    float tvar = (ts2 - (float)M3 * tmean * tmean) / (float)(M3 - 1);
    float h_temporal = 0.5f * logf(TWO_PI_E * (tvar + 1e-6f));

    // exact lower-median via 32-step binary search on ordered bits
    unsigned lo = 0u, hi = 0xFFFFFFFFu;
    for (int it = 0; it < 32; ++it) {
        unsigned mid = lo + ((hi - lo) >> 1);
        float cl = 0.f;
        for (int idx = tid; idx < TILE_ELEMS; idx += 256)
            cl += (f2ord(s_z[idx]) <= mid) ? 1.f : 0.f;
        float cnt = block_sum(cl, sred);
        if (cnt >= 4096.f) hi = mid; else lo = mid + 1u;
    }
    float med = ord2f(lo);

    float mean = vsum / (float)TILE_ELEMS;
    float var = (vsq - (float)TILE_ELEMS * mean * mean) / (float)(TILE_ELEMS - 1);
    float stdv = sqrtf(fmaxf(var, 0.f));
    float nrm = sqrtf(vsq);

    float h_score = score_sum / (float)TILE_ELEMS;
    float h_guid = -logf(1e-8f);  // var of a single scalar (population) is 0
    float m12 = d12 / (float)TILE_ELEMS;
    float m23 = d23 / (float)TILE_ELEMS;
    float h_solver = (logf(0.5f) + (0.01f + m12) / 0.005f - 0.5f) +
                     (logf(0.4f) + (0.0025f + m23) / 0.0008f - 0.5f);

    // feature tail: k_embed(64) | pos(5) | stats(10) | pad(1)
    float kf = (float)kptr[0];
    if (tid < 64) {
        int i = tid & 31;
        float freq = expf((float)i * (-logf(10000.f) / 31.f));
        float a = kf * freq;
        feat[(size_t)n * K1PAD + TILE_ELEMS + tid] = (tid < 32) ? sinf(a) : cosf(a);
    }
    if (tid == 0) {
        size_t base = (size_t)n * K1PAD + TILE_ELEMS + 64;
        float tf = (float)t, ff = (float)f;
        feat[base + 0] = sinf(0.1f * tf);
        feat[base + 1] = cosf(0.1f * tf);
        feat[base + 2] = sinf(0.1f * ff);
        feat[base + 3] = cosf(0.1f * ff);
        feat[base + 4] = 0.5f;  // S_SCALE / MAX_SCALES
        feat[base + 5] = mean;
        feat[base + 6] = stdv;
        feat[base + 7] = vmin;
        feat[base + 8] = vmax;
        feat[base + 9] = med;
        feat[base + 10] = var;
        feat[base + 11] = nrm;
        feat[base + 12] = sabs;
        feat[base + 13] = npos;
        feat[base + 14] = nneg;
        feat[base + 15] = 0.f;  // pad K to 8272
        comps[n * 8 + 0] = h_score;
        comps[n * 8 + 1] = h_guid;
        comps[n * 8 + 2] = h_solver;
        comps[n * 8 + 3] = h_temporal;
        comps[n * 8 + 4] = h_spec;
    }
}

// ---------------- pad W1 with one zero K-row ----------------
__global__ void pad_w1_kernel(const float* __restrict__ W1, float* __restrict__ W1p) {
    int i = blockIdx.x * blockDim.x + threadIdx.x;
    const int total = K1PAD * H1_DIM;
    if (i >= total) return;
    int row = i / H1_DIM;
    W1p[i] = (row < IN_DIM) ? W1[i] : 0.f;
}

// ---------------- one-wave WMMA f32 GEMM: out = relu(A @ W + bias) ----------------
// A: M x K (lda = K), W: K x ncols, out: M x ncols. One wave computes a 16x32
// output tile (two accumulators share each A fragment -> 2 WMMAs per K-step).
// Operand striping per CDNA5 ISA 7.12.2 (16x4 A / 4x16 B / 16x16 C-D, wave32).
__global__ void __launch_bounds__(32)
gemm16x32_wmma(const float* __restrict__ A, int lda,
               const float* __restrict__ W, int ncols,
               const float* __restrict__ bias,
               float* __restrict__ out, int K, int do_relu) {
    const int m0 = blockIdx.x * 16;
    const int n0 = blockIdx.y * 32;
    const int lane = threadIdx.x;
    const int half = lane >> 4;   // 0: lanes 0-15, 1: lanes 16-31
    const int l15 = lane & 15;

    const float* arow  = A + (size_t)(m0 + l15) * lda + 2 * half;  // K..K+1 / K+2..K+3
    const float* wcol0 = W + (size_t)(2 * half) * ncols + n0 + l15;
    const float* wcol1 = wcol0 + 16;

    __builtin_prefetch(arow, 0, 1);        // global_prefetch_b8
    __builtin_prefetch(wcol0, 0, 1);

    v8f c0 = {0.f, 0.f, 0.f, 0.f, 0.f, 0.f, 0.f, 0.f};
    v8f c1 = {0.f, 0.f, 0.f, 0.f, 0.f, 0.f, 0.f, 0.f};
    for (int k0 = 0; k0 < K; k0 += 4) {
        v2f a, b0, b1;
        a.x = arow[k0];
        a.y = arow[k0 + 1];
        b0.x = wcol0[(size_t)k0 * ncols];
        b0.y = wcol0[(size_t)(k0 + 1) * ncols];
        b1.x = wcol1[(size_t)k0 * ncols];
        b1.y = wcol1[(size_t)(k0 + 1) * ncols];
        c0 = __builtin_amdgcn_wmma_f32_16x16x4_f32(false, a, false, b0,
                                                   (short)0, c0, false, false);
        c1 = __builtin_amdgcn_wmma_f32_16x16x4_f32(false, a, false, b1,
                                                   (short)0, c1, false, false);
    }

#pragma unroll
    for (int r = 0; r < 8; ++r) {
        int row = m0 + r + half * 8;
        int col0 = n0 + l15;
        float v0 = c0[r] + bias[col0];
        float v1 = c1[r] + bias[col0 + 16];
        if (do_relu) { v0 = fmaxf(v0, 0.f); v1 = fmaxf(v1, 0.f); }
        out[(size_t)row * ncols + col0]      = v0;
        out[(size_t)row * ncols + col0 + 16] = v1;
    }
}

// ---------------- head: logits = H3 @ W4 + b4; softmax; dot with comps ----------------
__global__ void head_kernel(const float* __restrict__ H3,
                            const float* __restrict__ W4,
                            const float* __restrict__ b4,
                            const float* __restrict__ comps,
                            float* __restrict__ out, int N) {
    int n = blockIdx.x * blockDim.x + threadIdx.x;
    if (n >= N) return;
    float logits[5];
    for (int o = 0; o < 5; ++o) {
        float acc = b4[o];
        for (int i = 0; i < H3_DIM; ++i)
            acc += H3[(size_t)n * H3_DIM + i] * W4[i * 5 + o];
        logits[o] = acc;
    }
    float mx = logits[0];
    for (int o = 1; o < 5; ++o) mx = fmaxf(mx, logits[o]);
    float se = 0.f;
    for (int o = 0; o < 5; ++o) { logits[o] = expf(logits[o] - mx); se += logits[o]; }
    float inv = 1.f / se;
    float r = 0.f;
    for (int o = 0; o < 5; ++o) r += logits[o] * inv * comps[n * 8 + o];
    out[n] = r;
}

extern "C" void kernel_launch(void* const* d_in, const int* in_sizes, int n_in,
                              void* d_out, int out_size, void* d_ws, size_t ws_size,
                              hipStream_t stream) {
    (void)n_in; (void)out_size; (void)ws_size;
    const float* z   = (const float*)d_in[0];
    /* d_in[1] = y_obs: unused (guidance entropy is analytically constant) */
    const float* eh  = (const float*)d_in[2];
    const float* W1  = (const float*)d_in[3];
    const float* b1  = (const float*)d_in[4];
    const float* W2  = (const float*)d_in[5];
    const float* b2  = (const float*)d_in[6];
    const float* W3  = (const float*)d_in[7];
    const float* b3  = (const float*)d_in[8];
    const float* W4  = (const float*)d_in[9];
    const float* b4  = (const float*)d_in[10];
    const int* t_idx = (const int*)d_in[11];
    const int* f_idx = (const int*)d_in[12];
    const int* kptr  = (const int*)d_in[13];
    float* out = (float*)d_out;

    const int N = in_sizes[11];       // 512 nodes
    const int Neh = in_sizes[2] / 4;  // entropy_history is (4, N)

    // workspace layout (floats): feat | W1padded | comps | H1 | H2 | H3  (~26 MB)
    float* ws    = (float*)d_ws;
    float* feat  = ws;
    float* W1p   = feat + (size_t)N * K1PAD;
    float* comps = W1p + (size_t)K1PAD * H1_DIM;
    float* H1    = comps + (size_t)N * 8;
    float* H2    = H1 + (size_t)N * H1_DIM;
    float* H3    = H2 + (size_t)N * H2_DIM;

    {
        int total = K1PAD * H1_DIM;
        pad_w1_kernel<<<(total + 255) / 256, 256, 0, stream>>>(W1, W1p);
    }
    features_kernel<<<N, 256, 0, stream>>>(z, eh, Neh, t_idx, f_idx, kptr, feat, comps);
    gemm16x32_wmma<<<dim3(N / 16, H1_DIM / 32), 32, 0, stream>>>(feat, K1PAD, W1p, H1_DIM, b1, H1, K1PAD, 1);
    gemm16x32_wmma<<<dim3(N / 16, H2_DIM / 32), 32, 0, stream>>>(H1, H1_DIM, W2, H2_DIM, b2, H2, H1_DIM, 1);
    gemm16x32_wmma<<<dim3(N / 16, H3_DIM / 32), 32, 0, stream>>>(H2, H2_DIM, W3, H3_DIM, b3, H3, H2_DIM, 1);
    head_kernel<<<(N + 255) / 256, 256, 0, stream>>>(H3, W4, b4, comps, out, N);
}